// FF_56908316672276
// MI455X (gfx1250) — compile-verified
//
#include <hip/hip_runtime.h>
#include <hip/hip_bf16.h>

// Problem constants (from reference)
#define B_    32
#define T_    256
#define NIN_  128
#define H_    32
#define NOUT_ 128

#define IBLK  32               // rows of i handled per workgroup
#define NBLK  (T_ / IBLK)      // 8 row-blocks per batch

// LDS row strides (floats), padded for bank-conflict-free access (64 banks)
#define LDY  33                // y / h / o rows (33 odd -> conflict free)
#define LDA  260               // att rows (260 % 64 == 4 -> distinct banks over 16 rows)
#define LDC  36                // context rows

typedef float v2f __attribute__((ext_vector_type(2)));
typedef float v8f __attribute__((ext_vector_type(8)));

// Native CDNA5 tanh transcendental if the toolchain exposes it.
#if __has_builtin(__builtin_amdgcn_tanhf)
#define TANH_F32(x) __builtin_amdgcn_tanhf(x)
#else
#define TANH_F32(x) tanhf(x)
#endif

// Force a wave-uniform float into an SGPR (kills VGPR pressure for broadcasts).
static __device__ __forceinline__ float rfl(float x) {
    return __int_as_float(__builtin_amdgcn_readfirstlane(__float_as_int(x)));
}

// One fp32 WMMA step: D = A(16x4) * B(4x16) + C, exact fp32.
static __device__ __forceinline__ v8f wmma_f32(v2f a, v2f b, v8f c) {
    return __builtin_amdgcn_wmma_f32_16x16x4_f32(
        /*neg_a=*/false, a, /*neg_b=*/false, b,
        /*c_mod=*/(short)0, c, /*reuse_a=*/false, /*reuse_b=*/false);
}

__global__ __launch_bounds__(256)
void attn_fused_kernel(const float* __restrict__ x,     // [B,T,NIN]
                       const float* __restrict__ Win,   // [H,NIN]
                       const float* __restrict__ bin,   // [H]
                       const float* __restrict__ Wh,    // [H,H]
                       const float* __restrict__ Wo,    // [H,H]
                       const float* __restrict__ V,     // [H]
                       const float* __restrict__ Wout,  // [NOUT,H]
                       const float* __restrict__ bout,  // [NOUT]
                       float* __restrict__ out)         // [B,T,NOUT]
{
    __shared__ float sy[T_ * LDY];        // y  : 256 x 32 (padded)
    __shared__ float sh[T_ * LDY];        // h  : 256 x 32
    __shared__ float so[T_ * LDY];        // o  : 256 x 32
    __shared__ float satt[IBLK * LDA];    // att: 32 x 256 (padded)
    __shared__ float sctx[IBLK * LDC];    // ctx: 32 x 32

    const int blk   = blockIdx.x;
    const int b     = blk / NBLK;
    const int iBase = (blk % NBLK) * IBLK;
    const int tid   = threadIdx.x;
    const int lane  = tid & 31;
    const int wave  = tid >> 5;
    const int m16   = lane & 15;          // row/col index within 16
    const int half  = lane >> 4;          // 0: lanes 0-15, 1: lanes 16-31
    const int koff  = half * 2;           // K offset inside a 16x4 fragment

    const float* xb = x + (size_t)b * T_ * NIN_;

    // ---------------- Phase 1: y = x @ Win^T + bin  (M=256, N=32, K=128) ----
    // 16 M-tiles x 2 N-tiles = 32 tiles; 8 waves -> 4 tiles each.
    for (int mt = wave * 2; mt < wave * 2 + 2; ++mt) {
        const float* arow = xb + (mt * 16 + m16) * NIN_;
        for (int nt = 0; nt < 2; ++nt) {
            const float* brow = Win + (nt * 16 + m16) * NIN_;  // B = Win^T
            v8f acc = {};
            #pragma unroll 8
            for (int k0 = 0; k0 < NIN_; k0 += 4) {
                v2f a, bb;
                a.x  = arow[k0 + koff];
                a.y  = arow[k0 + koff + 1];
                bb.x = brow[k0 + koff];
                bb.y = brow[k0 + koff + 1];
                acc = wmma_f32(a, bb, acc);
            }
            const float bias = bin[nt * 16 + m16];
            #pragma unroll
            for (int v = 0; v < 8; ++v) {
                const int row = mt * 16 + v + half * 8;  // C layout
                sy[row * LDY + nt * 16 + m16] = acc[v] + bias;
            }
        }
    }
    __syncthreads();

    // ---------------- Phase 2: h = y @ Wh^T, o = y @ Wo^T  (M=256,N=32,K=32) -
    for (int which = 0; which < 2; ++which) {
        const float* Wm  = which ? Wo : Wh;
        float*       dst = which ? so : sh;
        for (int mt = wave * 2; mt < wave * 2 + 2; ++mt) {
            const int arow = (mt * 16 + m16) * LDY;
            for (int nt = 0; nt < 2; ++nt) {
                const float* brow = Wm + (nt * 16 + m16) * H_;
                v8f acc = {};
                #pragma unroll
                for (int k0 = 0; k0 < H_; k0 += 4) {
                    v2f a, bb;
                    a.x  = sy[arow + k0 + koff];
                    a.y  = sy[arow + k0 + koff + 1];
                    bb.x = brow[k0 + koff];
                    bb.y = brow[k0 + koff + 1];
                    acc = wmma_f32(a, bb, acc);
                }
                #pragma unroll
                for (int v = 0; v < 8; ++v) {
                    const int row = mt * 16 + v + half * 8;
                    dst[row * LDY + nt * 16 + m16] = acc[v];
                }
            }
        }
    }
    __syncthreads();

    // ---------------- Phase 3: e[i,j] = V . tanh(h_i + o_j); log_softmax ----
    // Each wave owns IBLK/8 = 4 rows i; each lane covers 8 j values.
    // V and h_i are wave-uniform -> pin them in SGPRs via readfirstlane.
    float vreg[H_];
    #pragma unroll
    for (int hh = 0; hh < H_; ++hh) vreg[hh] = rfl(V[hh]);

    for (int r = 0; r < IBLK / 8; ++r) {
        const int iLoc = wave * (IBLK / 8) + r;
        const int i    = iBase + iLoc;

        float hi[H_];
        #pragma unroll
        for (int hh = 0; hh < H_; ++hh) hi[hh] = rfl(sh[i * LDY + hh]);

        float e[8];
        float mx = -3.0e38f;
        #pragma unroll
        for (int jj = 0; jj < 8; ++jj) {
            const int j = jj * 32 + lane;
            const int jrow = j * LDY;
            float s = 0.f;
            #pragma unroll
            for (int hh = 0; hh < H_; ++hh)
                s += vreg[hh] * TANH_F32(hi[hh] + so[jrow + hh]);
            e[jj] = s;
            mx = fmaxf(mx, s);
        }
        // wave32 max reduction
        #pragma unroll
        for (int off = 16; off > 0; off >>= 1)
            mx = fmaxf(mx, __shfl_xor(mx, off, 32));
        float sum = 0.f;
        #pragma unroll
        for (int jj = 0; jj < 8; ++jj) sum += __expf(e[jj] - mx);
        #pragma unroll
        for (int off = 16; off > 0; off >>= 1)
            sum += __shfl_xor(sum, off, 32);
        const float lse = mx + __logf(sum);
        #pragma unroll
        for (int jj = 0; jj < 8; ++jj)
            satt[iLoc * LDA + jj * 32 + lane] = e[jj] - lse;  // log_softmax (faithful)
    }
    __syncthreads();

    // ---------------- Phase 4: context = att @ y  (M=32, N=32, K=256) -------
    if (wave < 4) {
        const int mt = wave >> 1;
        const int nt = wave & 1;
        const int arow = (mt * 16 + m16) * LDA;
        v8f acc = {};
        #pragma unroll 8
        for (int k0 = 0; k0 < T_; k0 += 4) {
            v2f a, bb;
            a.x  = satt[arow + k0 + koff];
            a.y  = satt[arow + k0 + koff + 1];
            bb.x = sy[(k0 + koff)     * LDY + nt * 16 + m16];  // B = y (K x N)
            bb.y = sy[(k0 + koff + 1) * LDY + nt * 16 + m16];
            acc = wmma_f32(a, bb, acc);
        }
        #pragma unroll
        for (int v = 0; v < 8; ++v) {
            const int row = mt * 16 + v + half * 8;
            sctx[row * LDC + nt * 16 + m16] = acc[v];
        }
    }
    __syncthreads();

    // ---------------- Phase 5: y_hat = context @ Wout^T + bout  (32x128,K=32)
    for (int t = wave; t < 16; t += 8) {
        const int mt = t >> 3;        // 0..1
        const int nt = t & 7;         // 0..7
        const int arow = (mt * 16 + m16) * LDC;
        const float* brow = Wout + (nt * 16 + m16) * H_;  // B = Wout^T
        v8f acc = {};
        #pragma unroll
        for (int k0 = 0; k0 < H_; k0 += 4) {
            v2f a, bb;
            a.x  = sctx[arow + k0 + koff];
            a.y  = sctx[arow + k0 + koff + 1];
            bb.x = brow[k0 + koff];
            bb.y = brow[k0 + koff + 1];
            acc = wmma_f32(a, bb, acc);
        }
        const float bias = bout[nt * 16 + m16];
        #pragma unroll
        for (int v = 0; v < 8; ++v) {
            const int row = iBase + mt * 16 + v + half * 8;
            out[((size_t)b * T_ + row) * NOUT_ + nt * 16 + m16] = acc[v] + bias;
        }
    }
}

extern "C" void kernel_launch(void* const* d_in, const int* in_sizes, int n_in,
                              void* d_out, int out_size, void* d_ws, size_t ws_size,
                              hipStream_t stream) {
    (void)in_sizes; (void)n_in; (void)out_size; (void)d_ws; (void)ws_size;
    const float* x    = (const float*)d_in[0];
    const float* Win  = (const float*)d_in[1];
    const float* bin  = (const float*)d_in[2];
    const float* Wh   = (const float*)d_in[3];
    const float* Wo   = (const float*)d_in[4];
    const float* V    = (const float*)d_in[5];
    const float* Wout = (const float*)d_in[6];
    const float* bout = (const float*)d_in[7];
    // d_in[8] = batch_size scalar (compile-time constant here)
    float* out = (float*)d_out;

    dim3 grid(B_ * NBLK);   // 256 workgroups
    dim3 block(256);        // 8 wave32 per workgroup
    attn_fused_kernel<<<grid, block, 0, stream>>>(x, Win, bin, Wh, Wo, V,
                                                  Wout, bout, out);
}